// FraudGCN_71803263254611
// MI455X (gfx1250) — compile-verified
//
#include <hip/hip_runtime.h>
#include <hip/hip_bf16.h>

typedef float v2f __attribute__((ext_vector_type(2)));
typedef float v8f __attribute__((ext_vector_type(8)));

__device__ __forceinline__ void atomAddF(float* p, float v) {
    // relaxed, device scope -> native global_atomic_add_f32 on gfx1250
    __hip_atomic_fetch_add(p, v, __ATOMIC_RELAXED, __HIP_MEMORY_SCOPE_AGENT);
}

// ---------------- utility kernels ----------------
__global__ void fill_kernel(float* __restrict__ p, float v, int n) {
    int t = blockIdx.x * blockDim.x + threadIdx.x;
    if (t < n) p[t] = v;
}

__global__ void deg_kernel(const int* __restrict__ dst, float* __restrict__ deg, int ne) {
    int e = blockIdx.x * blockDim.x + threadIdx.x;
    if (e < ne) atomAddF(&deg[dst[e]], 1.0f);
}

__global__ void rsqrt_kernel(float* __restrict__ d, int n) {
    int t = blockIdx.x * blockDim.x + threadIdx.x;
    if (t < n) d[t] = rsqrtf(d[t]);   // deg >= 1 always (self loops)
}

// ---------------- layer 1 GEMM: h1[N,32] = x[N,7] @ W[7,32] (fp32 WMMA) ----------------
// One wave per 16-row tile; only full tiles (ntile = n/16). No guards -> clean WMMA codegen.
__global__ void gemm1_wmma(const float* __restrict__ x, const float* __restrict__ W,
                           float* __restrict__ h, int ntile) {
    int wave = (blockIdx.x * blockDim.x + threadIdx.x) >> 5;
    int lane = threadIdx.x & 31;
    if (wave >= ntile) return;             // wave-uniform: EXEC stays all-1s
    int hs = lane >> 4;                    // half-select (lanes 16-31 hold K+2)
    int l  = lane & 15;
    int kb = hs * 2;
    float zmask = hs ? 0.0f : 1.0f;        // zero out padded K=7 lane via value select
    const float* xr = x + (size_t)(wave * 16 + l) * 7;

    // unconditional in-range loads; select on value (no divergent load blocks)
    float x5  = xr[5];
    v2f a0, a1;
    a0[0] = xr[kb + 0]; a0[1] = xr[kb + 1];            // K = kb, kb+1
    a1[0] = xr[kb + 4]; a1[1] = x5 * zmask;            // K = kb+4, kb+5 (K=7 padded w/ 0)

    float w5a = W[5 * 32 + l];
    float w5b = W[5 * 32 + 16 + l];

    v8f c0 = {}, c1 = {};
    v2f b0, b1;
    // column half 0 (cols 0..15)
    b0[0] = W[(kb + 0) * 32 + l];  b0[1] = W[(kb + 1) * 32 + l];
    b1[0] = W[(kb + 4) * 32 + l];  b1[1] = w5a * zmask;
    c0 = __builtin_amdgcn_wmma_f32_16x16x4_f32(false, a0, false, b0, (short)0, c0, false, false);
    c0 = __builtin_amdgcn_wmma_f32_16x16x4_f32(false, a1, false, b1, (short)0, c0, false, false);
    // column half 1 (cols 16..31)
    b0[0] = W[(kb + 0) * 32 + 16 + l];  b0[1] = W[(kb + 1) * 32 + 16 + l];
    b1[0] = W[(kb + 4) * 32 + 16 + l];  b1[1] = w5b * zmask;
    c1 = __builtin_amdgcn_wmma_f32_16x16x4_f32(false, a0, false, b0, (short)0, c1, false, false);
    c1 = __builtin_amdgcn_wmma_f32_16x16x4_f32(false, a1, false, b1, (short)0, c1, false, false);

    int orow = wave * 16 + hs * 8;         // C layout: VGPR r -> row r (lo lanes) / r+8 (hi lanes)
    float* hp = h + (size_t)orow * 32 + l;
#pragma unroll
    for (int r = 0; r < 8; ++r) {
        hp[(size_t)r * 32]      = c0[r];
        hp[(size_t)r * 32 + 16] = c1[r];
    }
}

// ---------------- layer 2 GEMM: h2[N,16] = y1[N,32] @ W[32,16] (fp32 WMMA) ----------------
__global__ void gemm2_wmma(const float* __restrict__ y, const float* __restrict__ W,
                           float* __restrict__ h, int ntile) {
    int wave = (blockIdx.x * blockDim.x + threadIdx.x) >> 5;
    int lane = threadIdx.x & 31;
    if (wave >= ntile) return;
    int hs = lane >> 4;
    int l  = lane & 15;
    const float* yr = y + (size_t)(wave * 16 + l) * 32;

    v8f c = {};
#pragma unroll
    for (int kk = 0; kk < 8; ++kk) {       // K = 32 in steps of 4
        int k0 = kk * 4 + hs * 2;
        v2f a = *(const v2f*)(yr + k0);    // 8B-aligned (k0 even, row base 128B-aligned)
        v2f b;
        b[0] = W[k0 * 16 + l];   b[1] = W[(k0 + 1) * 16 + l];
        c = __builtin_amdgcn_wmma_f32_16x16x4_f32(false, a, false, b, (short)0, c, false, false);
    }
    int orow = wave * 16 + hs * 8;
    float* hp = h + (size_t)orow * 16 + l;
#pragma unroll
    for (int r = 0; r < 8; ++r) hp[(size_t)r * 16] = c[r];
}

// ---------------- scalar tail for rows not covered by full 16-row tiles ----------------
__global__ void gemm_tail(const float* __restrict__ x, const float* __restrict__ W,
                          float* __restrict__ h, int row0, int n, int kin, int kout) {
    int t = blockIdx.x * blockDim.x + threadIdx.x;
    int nrem = (n - row0) * kout;
    if (t >= nrem) return;
    int row = row0 + t / kout;
    int c   = t % kout;
    float acc = 0.0f;
    for (int k = 0; k < kin; ++k) acc += x[(size_t)row * kin + k] * W[(size_t)k * kout + c];
    h[(size_t)row * kout + c] = acc;
}

// ---------------- edge scatter-add: agg[dst] += h[src] * dinv[src]*dinv[dst] ----------------
template <int W>   // feature width (32 or 16); W/8 threads per edge, 8 cols each
__global__ void edge_agg(const int* __restrict__ src, const int* __restrict__ dst,
                         const float* __restrict__ h, const float* __restrict__ dinv,
                         float* __restrict__ agg, int ne) {
    constexpr int P = W / 8;
    int t = blockIdx.x * blockDim.x + threadIdx.x;
    if (t >= ne * P) return;
    int e    = t / P;
    int part = t - e * P;
    int s = src[e];
    int d = dst[e];
    float w = dinv[s] * dinv[d];
    const float* hp = h + (size_t)s * W + part * 8;
    float4 u0 = *(const float4*)(hp);
    float4 u1 = *(const float4*)(hp + 4);
    float* ap = agg + (size_t)d * W + part * 8;
    atomAddF(ap + 0, u0.x * w);  atomAddF(ap + 1, u0.y * w);
    atomAddF(ap + 2, u0.z * w);  atomAddF(ap + 3, u0.w * w);
    atomAddF(ap + 4, u1.x * w);  atomAddF(ap + 5, u1.y * w);
    atomAddF(ap + 6, u1.z * w);  atomAddF(ap + 7, u1.w * w);
}

// ---------------- self-loop + bias + ReLU, written in place over h ----------------
template <int W>
__global__ void finalize(float* __restrict__ h, const float* __restrict__ agg,
                         const float* __restrict__ dinv, const float* __restrict__ bias, int n) {
    int t = blockIdx.x * blockDim.x + threadIdx.x;
    if (t >= n * W) return;
    int node = t / W;
    int j    = t - node * W;
    float di = dinv[node];
    float v  = agg[t] + h[t] * di * di + bias[j];
    h[t] = v > 0.0f ? v : 0.0f;
}

// ---------------- final FC: out[N,2] = y2[N,16] @ Wfc[16,2] + bfc ----------------
__global__ void fc_kernel(const float* __restrict__ y, const float* __restrict__ Wfc,
                          const float* __restrict__ bfc, float* __restrict__ out, int n) {
    int i = blockIdx.x * blockDim.x + threadIdx.x;
    if (i >= n) return;
    const float* yr = y + (size_t)i * 16;
    float a0 = bfc[0], a1 = bfc[1];
#pragma unroll
    for (int j = 0; j < 16; ++j) {
        float v = yr[j];
        a0 += v * Wfc[j * 2 + 0];
        a1 += v * Wfc[j * 2 + 1];
    }
    out[(size_t)i * 2 + 0] = a0;
    out[(size_t)i * 2 + 1] = a1;
}

static inline unsigned nblk(long long n, int b) { return (unsigned)((n + b - 1) / b); }

extern "C" void kernel_launch(void* const* d_in, const int* in_sizes, int n_in,
                              void* d_out, int out_size, void* d_ws, size_t ws_size,
                              hipStream_t stream) {
    const float* x   = (const float*)d_in[0];
    const int*   ei  = (const int*)d_in[1];
    const float* W1  = (const float*)d_in[2];
    const float* b1  = (const float*)d_in[3];
    const float* W2  = (const float*)d_in[4];
    const float* b2  = (const float*)d_in[5];
    const float* Wfc = (const float*)d_in[6];
    const float* bfc = (const float*)d_in[7];

    const int N = in_sizes[0] / 7;      // 200000
    const int E = in_sizes[1] / 2;      // 6400000
    const int* src = ei;
    const int* dst = ei + E;

    float* ws   = (float*)d_ws;
    float* dinv = ws;                                   // N
    float* h1   = ws + (size_t)N;                       // 32N (becomes y1 in place)
    float* agg1 = ws + (size_t)33 * N;                  // 32N
    float* h2   = ws + (size_t)33 * N;                  // 16N (reuses agg1 after layer 1)
    float* agg2 = ws + (size_t)49 * N;                  // 16N
    float* out  = (float*)d_out;

    const int B     = 256;
    const int ntile = N / 16;                           // full tiles only
    const int row0  = ntile * 16;
    const int rem   = N - row0;                         // 0 for N=200000

    // degree -> dinv
    fill_kernel<<<nblk(N, B), B, 0, stream>>>(dinv, 1.0f, N);
    deg_kernel<<<nblk(E, B), B, 0, stream>>>(dst, dinv, E);
    rsqrt_kernel<<<nblk(N, B), B, 0, stream>>>(dinv, N);

    // layer 1
    gemm1_wmma<<<nblk((long long)ntile * 32, B), B, 0, stream>>>(x, W1, h1, ntile);
    if (rem > 0)
        gemm_tail<<<nblk((long long)rem * 32, B), B, 0, stream>>>(x, W1, h1, row0, N, 7, 32);
    fill_kernel<<<nblk((long long)N * 32, B), B, 0, stream>>>(agg1, 0.0f, N * 32);
    edge_agg<32><<<nblk((long long)E * 4, B), B, 0, stream>>>(src, dst, h1, dinv, agg1, E);
    finalize<32><<<nblk((long long)N * 32, B), B, 0, stream>>>(h1, agg1, dinv, b1, N);

    // layer 2
    gemm2_wmma<<<nblk((long long)ntile * 32, B), B, 0, stream>>>(h1, W2, h2, ntile);
    if (rem > 0)
        gemm_tail<<<nblk((long long)rem * 16, B), B, 0, stream>>>(h1, W2, h2, row0, N, 32, 16);
    fill_kernel<<<nblk((long long)N * 16, B), B, 0, stream>>>(agg2, 0.0f, N * 16);
    edge_agg<16><<<nblk((long long)E * 2, B), B, 0, stream>>>(src, dst, h2, dinv, agg2, E);
    finalize<16><<<nblk((long long)N * 16, B), B, 0, stream>>>(h2, agg2, dinv, b2, N);

    // FC head
    fc_kernel<<<nblk(N, B), B, 0, stream>>>(h2, Wfc, bfc, out, N);
}